// IntSoftmax_31688268710294
// MI455X (gfx1250) — compile-verified
//
#include <hip/hip_runtime.h>

// IntSoftmax over rows of 1024 fp32, shape (4,12,1024,1024).
// One wave32 per row, row resident in 32 VGPRs/lane, async global->LDS
// double-buffered staging (ASYNCcnt), shfl_xor reductions, NT b128 stores.

#define ROW_LEN   1024
#define WAVE      32
#define WPB       8                 // waves per block
#define BLOCK     (WAVE * WPB)      // 256 threads = 8 wave32
#define CHUNKS    8                 // 8 x float4 per lane = 32 floats
#define ROW_BYTES (ROW_LEN * 4)     // 4096 B per row buffer
#define ROWS_PER_WAVE 4

typedef float vf4 __attribute__((ext_vector_type(4)));
typedef __attribute__((address_space(3))) unsigned char lds_byte;

// Per-lane async copy of one 4KB row into LDS: 8 x global_load_async_to_lds_b128
// (each lane moves 16B per chunk; one chunk = 512B contiguous). Tracked by ASYNCcnt.
__device__ __forceinline__ void row_async_load(const float* __restrict__ gsrc,
                                               unsigned lds_base_bytes, int lane) {
#pragma unroll
  for (int k = 0; k < CHUNKS; ++k) {
    unsigned dst = lds_base_bytes + (unsigned)(k * 512 + lane * 16);
    const float* src = gsrc + k * 128 + lane * 4;
    asm volatile("global_load_async_to_lds_b128 %0, %1, off"
                 :: "v"(dst), "v"(src)
                 : "memory");
  }
}

__global__ __launch_bounds__(BLOCK) void intsoftmax_gfx1250(
    const float* __restrict__ x, const float* __restrict__ scale,
    float* __restrict__ out, int rows) {
  __shared__ __align__(16) unsigned char smem[WPB * 2 * ROW_BYTES];

  const int lane = threadIdx.x & (WAVE - 1);
  const int wid  = threadIdx.x >> 5;
  const int gw   = blockIdx.x * WPB + wid;     // global wave id (one row at a time)
  const int nw   = gridDim.x * WPB;            // total waves

  // LDS byte offset of this wave's staging area (AS3 offset value, which is
  // exactly what the async-to-LDS instruction consumes in its dst VGPR).
  const unsigned lds_wave =
      (unsigned)(unsigned long long)((lds_byte*)smem) + (unsigned)(wid * 2 * ROW_BYTES);

  // Wave-uniform constants (match reference evaluation in fp32).
  const float s       = scale[0];
  const float inv_s   = 1.0f / s;              // x/s -> x*inv_s (monotone)
  const float x0      = floorf(-1.0f / s);     // x0 = floor(-1/s)  (fp32: -50 for s=0.02)
  const float inv_x0  = 1.0f / x0;
  const float clampv  = 15.0f * x0;            // N_SHIFT * x0
  const float INTMAXF = 2147483647.0f;

  if (gw >= rows) return;                      // wave-uniform exit

  int buf = 0;
  row_async_load(x + (size_t)gw * ROW_LEN, lds_wave, lane);   // prologue prefetch

  for (int row = gw; row < rows; row += nw) {
    const int next = row + nw;
    if (next < rows) {
      // Prefetch next row into the other buffer, then wait only for the
      // current row's 8 loads (async loads complete in order).
      row_async_load(x + (size_t)next * ROW_LEN,
                     lds_wave + (unsigned)((buf ^ 1) * ROW_BYTES), lane);
      asm volatile("s_wait_asynccnt 0x8" ::: "memory");
    } else {
      asm volatile("s_wait_asynccnt 0x0" ::: "memory");
    }

    // ---- Pass 1: LDS -> registers, dequantize, rowwise max ----
    const vf4* lp = (const vf4*)(smem + wid * 2 * ROW_BYTES + buf * ROW_BYTES);
    vf4 d[CHUNKS];
    float m = -3.402823466e38f;
#pragma unroll
    for (int k = 0; k < CHUNKS; ++k) {
      vf4 v = lp[k * 32 + lane];               // ds_load_b128
#pragma unroll
      for (int j = 0; j < 4; ++j) {
        float xi = v[j] * inv_s;               // x_int = x / s
        v[j] = xi;
        m = fmaxf(m, xi);
      }
      d[k] = v;
    }
#pragma unroll
    for (int off = 16; off > 0; off >>= 1)     // wave32 max reduction
      m = fmaxf(m, __shfl_xor(m, off, WAVE));

    // ---- Pass 2: integer-exp approximation + rowwise sum (registers only) ----
    float sum = 0.0f;
#pragma unroll
    for (int k = 0; k < CHUNKS; ++k) {
#pragma unroll
      for (int j = 0; j < 4; ++j) {
        float t = d[k][j] - m;
        t = t + floorf(t * 0.5f) - floorf(t * 0.0625f);   // x*(1+1/2-1/16)
        t = fmaxf(t, clampv);                             // max(x, 15*x0)
        float q = floorf(t * inv_x0);                     // q = floor(x/x0)
        float r = fmaf(-x0, q, t);                        // r = x - x0*q
        float e = fmaf(r, 0.5f, -x0);                     // r/2 - x0
        e = floorf(ldexpf(e, 15 - (int)q));               // floor(e * 2^(15-q))
        e = fmaxf(e, 0.0f);
        d[k][j] = e;
        sum += e;
      }
    }
#pragma unroll
    for (int off = 16; off > 0; off >>= 1)     // wave32 sum reduction
      sum += __shfl_xor(sum, off, WAVE);
    sum = fminf(sum, INTMAXF);
    const float factor = floorf(INTMAXF / sum);

    // ---- Pass 3: normalize, requantize, stream out (non-temporal b128) ----
    float* orow = out + (size_t)row * ROW_LEN;
#pragma unroll
    for (int k = 0; k < CHUNKS; ++k) {
      vf4 o;
#pragma unroll
      for (int j = 0; j < 4; ++j) {
        float v = floorf(d[k][j] * factor * (1.0f / 16777216.0f)); // /2^(31-8+1)
        o[j] = v * (1.0f / 128.0f);                                // * 2^-(OUTPUT_BIT-1)
      }
      __builtin_nontemporal_store(o, (vf4*)orow + k * 32 + lane);
    }
    buf ^= 1;
  }
}

extern "C" void kernel_launch(void* const* d_in, const int* in_sizes, int n_in,
                              void* d_out, int out_size, void* d_ws, size_t ws_size,
                              hipStream_t stream) {
  (void)n_in; (void)out_size; (void)d_ws; (void)ws_size;
  const float* x     = (const float*)d_in[0];
  const float* scale = (const float*)d_in[1];
  float* out         = (float*)d_out;

  const int rows = in_sizes[0] / ROW_LEN;                        // 49152
  const int total_waves = (rows + ROWS_PER_WAVE - 1) / ROWS_PER_WAVE;
  const int blocks = (total_waves + WPB - 1) / WPB;              // 1536

  intsoftmax_gfx1250<<<blocks, BLOCK, 0, stream>>>(x, scale, out, rows);
}